// LIFEncoder_6476810682689
// MI455X (gfx1250) — compile-verified
//
#include <hip/hip_runtime.h>
#include <hip/hip_bf16.h>
#include <stdint.h>

// ---- LIF constants (from reference) ----
#define NB_STEPS   100
#define F          512            // raw features
#define FE         1024           // features * POP_SIZE
#define ALPHA      0.8187307530779818f   // exp(-1/5)
#define BETA       0.9048374180359595f   // exp(-1/10)
#define GAIN       0.05f
#define SHIFT      4.0f
#define THRESH     1.0f
#define REFSTEPS   5.0f
// noise = (sum4bytes - 510) * NOISE_SCALE ; std(sum4 U{0..255}) = sqrt(4*(65536-1)/12) = 147.80
// NOISE_SCALE = 0.3 / 147.80
#define NOISE_SCALE 2.0297692e-3f

typedef __attribute__((ext_vector_type(4))) float f32x4;

__device__ __forceinline__ uint32_t mix32(uint32_t x) {
    x ^= x >> 16; x *= 0x7feb352du;
    x ^= x >> 15; x *= 0x846ca68bu;
    x ^= x >> 16;
    return x;
}

__device__ __forceinline__ uint32_t bytesum(uint32_t h) {
#if __has_builtin(__builtin_amdgcn_udot4)
    // v_dot4_u32_u8: sum of the 4 bytes in one VALU op
    return __builtin_amdgcn_udot4(h, 0x01010101u, 0u, false);
#else
    return (h & 0xffu) + ((h >> 8) & 0xffu) + ((h >> 16) & 0xffu) + (h >> 24);
#endif
}

// One block per sample; each thread owns 4 consecutive encoder neurons.
__global__ __launch_bounds__(256) void lif_encode_kernel(
        const float* __restrict__ X, float* __restrict__ out) {
    __shared__ float xs[F];   // one input row (2 KB)

    const int n   = blockIdx.x;
    const int tid = threadIdx.x;      // 0..255

    // ---- CDNA5 async copy: stage X[n][0:512] into LDS, 8B per lane ----
    {
        uint64_t gaddr = (uint64_t)(uintptr_t)(X + (size_t)n * F + 2 * tid);
        uint32_t laddr = (uint32_t)(uintptr_t)xs + 8u * (uint32_t)tid;
        asm volatile("global_load_async_to_lds_b64 %0, %1, off"
                     :: "v"(laddr), "v"(gaddr) : "memory");
        asm volatile("s_wait_asynccnt 0" ::: "memory");
    }
    __syncthreads();

    // ---- per-thread state: 4 neurons (fe = 4*tid + j ; feature = fe>>1) ----
    float Iin[4], syn[4], mem[4], refr[4];
    uint32_t rs[4];
    const uint32_t gbase = (uint32_t)n * FE + 4u * (uint32_t)tid;
#pragma unroll
    for (int j = 0; j < 4; ++j) {
        float xv = xs[2 * tid + (j >> 1)];          // repeat(POP_SIZE=2)
        Iin[j]  = (xv + SHIFT) * GAIN;
        uint32_t gid = gbase + (uint32_t)j;
        // mem0 ~ U[0, 0.5)
        mem[j]  = (float)mix32(gid ^ 0xC2B2AE35u) * (0.5f / 4294967296.0f);
        syn[j]  = 0.0f;
        refr[j] = 0.0f;
        rs[j]   = mix32(gid * 0x9E3779B9u + 1u) | 1u;   // per-neuron LCG stream
    }

    float* op = out + (size_t)n * (NB_STEPS * FE) + 4u * (uint32_t)tid;

#pragma unroll 2
    for (int t = 0; t < NB_STEPS; ++t) {
        float spk[4];
#pragma unroll
        for (int j = 0; j < 4; ++j) {
            // cheap per-step Gaussian approx: LCG -> mix -> Irwin-Hall(4 bytes)
            rs[j] = rs[j] * 1664525u + 1013904223u;
            uint32_t h = rs[j] ^ (rs[j] >> 16);
            float noise = ((float)(int)bytesum(h) - 510.0f) * NOISE_SCALE;

            syn[j] = ALPHA * syn[j] + (Iin[j] + noise);
            mem[j] = BETA * mem[j] + syn[j];
            float mem_eff = (refr[j] <= 0.0f) ? mem[j] : 0.0f;
            float s = (mem_eff > THRESH) ? 1.0f : 0.0f;
            mem[j]  = mem_eff - s * THRESH;
            refr[j] = fmaxf(refr[j] - 1.0f, 0.0f) + s * REFSTEPS;
            spk[j] = s;
        }
        f32x4 v = { spk[0], spk[1], spk[2], spk[3] };
        // 839 MB of write-once data: non-temporal b128 store
        __builtin_nontemporal_store(v, (f32x4*)op);
        op += FE;
    }
}

extern "C" void kernel_launch(void* const* d_in, const int* in_sizes, int n_in,
                              void* d_out, int out_size, void* d_ws, size_t ws_size,
                              hipStream_t stream) {
    const float* X = (const float*)d_in[0];
    float* out = (float*)d_out;
    const int n_samples = in_sizes[0] / F;        // 2048
    dim3 grid((unsigned)n_samples), block(256);
    hipLaunchKernelGGL(lif_encode_kernel, grid, block, 0, stream, X, out);
}